// SAGPool_22299470201293
// MI455X (gfx1250) — compile-verified
//
#include <hip/hip_runtime.h>
#include <hip/hip_bf16.h>

#define N_NODES   100000
#define N_EDGES   3200000
#define DIM       128
#define USERS     5
#define KSEL      79996              /* ceil(0.8*(N-USERS)) */
#define PERM_LEN  (KSEL + USERS)     /* 80001 */
#define NEG_SLOPE 0.2f
#define N_TILES   (N_NODES / 16)     /* 6250, exact */

/* ---- d_out layout (floats, concatenated in reference return order) ---- */
#define X_OFF     0
#define EI_OFF    (PERM_LEN * DIM)                   /* 10,240,128 */
#define EA_OFF    (EI_OFF + 2 * N_EDGES)             /* 16,640,128 */
#define BATCH_OFF (EA_OFF + N_EDGES)                 /* 19,840,128 */
#define PERM_OFF  (BATCH_OFF + PERM_LEN)
#define NID_OFF   (PERM_OFF + PERM_LEN)

/* ---- workspace layout (dword units) ---- */
#define WS_XL     0
#define WS_XR     (N_NODES)
#define WS_AMAX   (2 * N_NODES)
#define WS_DENOM  (3 * N_NODES)
#define WS_NUM    (4 * N_NODES)
#define WS_SCORE  (5 * N_NODES)
#define WS_KEYS   (6 * N_NODES)
#define WS_MAP    (7 * N_NODES)
#define WS_PERM   (8 * N_NODES)
#define WS_CTR    (8 * N_NODES + PERM_LEN)

typedef __attribute__((ext_vector_type(2))) float v2f;
typedef __attribute__((ext_vector_type(8))) float v8f;

/* order-preserving float<->uint encoding for atomic max over floats */
__device__ __forceinline__ unsigned encodeF(float f) {
    unsigned u = __float_as_uint(f);
    return (u & 0x80000000u) ? ~u : (u | 0x80000000u);
}
__device__ __forceinline__ float decodeF(unsigned u) {
    u = (u & 0x80000000u) ? (u & 0x7FFFFFFFu) : ~u;
    return __uint_as_float(u);
}

/* LDS byte offset of a generic pointer to a __shared__ object:
   per ISA, generic LDS addresses carry the wave-relative offset in addr[31:0]. */
__device__ __forceinline__ unsigned lds_offset(const void* p) {
    return (unsigned)(unsigned long long)p;
}

/* -------- K0: per-node init -------- */
__global__ void k_init(unsigned* __restrict__ amaxU, float* __restrict__ denom,
                       float* __restrict__ num, int* __restrict__ mapping) {
    int n = blockIdx.x * blockDim.x + threadIdx.x;
    if (n >= N_NODES) return;
    amaxU[n] = 0u;      /* identity of the ordered-uint encoding */
    denom[n] = 0.f;
    num[n]   = 0.f;
    mapping[n] = -1;
}

/* -------- K1: x_l = x@W_l + b_l, x_r = x@W_r + b_r via V_WMMA_F32_16X16X4_F32.
   Block = 8 waves; the block's 128x128 fp32 x-tile (64 KB) is staged into LDS with
   async global->LDS B128 copies (coalesced), fenced by s_wait_asynccnt + barrier,
   then each wave feeds its 16-row tile to WMMA from swizzled LDS (ds_load_b64,
   bank-conflict-free across the 16 lanes). B = [128x16], col0 = W_l, col1 = W_r. */
__global__ __launch_bounds__(256) void k_gemv_wmma(
    const float* __restrict__ x, const float* __restrict__ Wl,
    const float* __restrict__ Wr, const float* __restrict__ bl,
    const float* __restrict__ br, float* __restrict__ xl, float* __restrict__ xr)
{
    __shared__ float smem[128 * DIM];            /* 64 KB of the 320 KB WGP LDS */
    const int tid       = threadIdx.x;
    const int blockRow0 = blockIdx.x * 128;

    /* ---- async stage: 4096 float4 chunks, 16 per thread ---- */
    #pragma unroll
    for (int it = 0; it < 16; ++it) {
        int i    = it * 256 + tid;               /* chunk id 0..4095   */
        int r    = i >> 5;                       /* tile row 0..127    */
        int c    = (i & 31) << 2;                /* col base 0..124    */
        int grow = blockRow0 + r;
        if (grow < N_NODES) {
            const float* gp = x + (unsigned long long)grow * DIM + c;
            unsigned loff = lds_offset(&smem[r * DIM + (c ^ ((r & 15) << 2))]);
            asm volatile("global_load_async_to_lds_b128 %0, %1, off"
                         :: "v"(loff), "v"(gp) : "memory");
        }
    }
    asm volatile("s_wait_asynccnt 0x0" ::: "memory");
    __syncthreads();

    const int lane    = tid & 31;
    const int wave    = tid >> 5;
    const int tileId  = blockIdx.x * 8 + wave;
    if (tileId >= N_TILES) return;               /* wave-uniform: EXEC stays all-ones */

    const int l15      = lane & 15;
    const int halfsel  = lane >> 4;              /* 0: K pair {0,1}; 1: K pair {2,3} */
    const int rowLocal = wave * 16 + l15;
    const int swz      = l15 << 2;

    v8f c = {};
    #pragma unroll 4
    for (int kb = 0; kb < DIM; kb += 4) {
        const int ka = kb + (halfsel << 1);
        const float* ap = &smem[rowLocal * DIM + (ka ^ swz)];
        v2f a;
        a.x = ap[0];
        a.y = ap[1];
        v2f b;
        float b0 = 0.f, b1 = 0.f;
        if (l15 == 0)      { b0 = Wl[ka]; b1 = Wl[ka + 1]; }
        else if (l15 == 1) { b0 = Wr[ka]; b1 = Wr[ka + 1]; }
        b.x = b0; b.y = b1;
        /* 8-arg pattern: (neg_a, A, neg_b, B, c_mod, C, reuse_a, reuse_b) */
        c = __builtin_amdgcn_wmma_f32_16x16x4_f32(false, a, false, b,
                                                  (short)0, c, false, false);
    }

    /* C layout: lane l holds (M = v + 8*(l>>4), N = l&15); N=0 -> x_l, N=1 -> x_r */
    if (l15 < 2) {
        float* outp = (l15 == 0) ? xl : xr;
        const float badd = (l15 == 0) ? bl[0] : br[0];
        const int mbase = tileId * 16 + (halfsel << 3);
        #pragma unroll
        for (int v = 0; v < 8; ++v)
            outp[mbase + v] = c[v] + badd;
    }
}

/* -------- K2: per-edge (incl. self-loops) segment max of alpha over dst -------- */
__global__ void k_edge_max(const int* __restrict__ ei, const float* __restrict__ xl,
                           const float* __restrict__ xr, const float* __restrict__ att,
                           unsigned* __restrict__ amaxU)
{
    int e = blockIdx.x * blockDim.x + threadIdx.x;
    if (e >= N_EDGES + N_NODES) return;
    __builtin_prefetch(ei + e + 8192, 0, 1);   /* global_prefetch_b8 */
    int s, d;
    if (e < N_EDGES) { s = __builtin_nontemporal_load(ei + e);
                       d = __builtin_nontemporal_load(ei + N_EDGES + e); }
    else             { s = e - N_EDGES; d = s; }
    float a = xl[s] + xr[d];
    a = (a > 0.f) ? a : NEG_SLOPE * a;
    atomicMax(amaxU + d, encodeF(a * att[0]));
}

/* -------- K3: per-edge exp + fused num/denom accumulation -------- */
__global__ void k_edge_sum(const int* __restrict__ ei, const float* __restrict__ xl,
                           const float* __restrict__ xr, const float* __restrict__ att,
                           const unsigned* __restrict__ amaxU,
                           float* __restrict__ denom, float* __restrict__ num)
{
    int e = blockIdx.x * blockDim.x + threadIdx.x;
    if (e >= N_EDGES + N_NODES) return;
    __builtin_prefetch(ei + e + 8192, 0, 1);
    int s, d;
    if (e < N_EDGES) { s = __builtin_nontemporal_load(ei + e);
                       d = __builtin_nontemporal_load(ei + N_EDGES + e); }
    else             { s = e - N_EDGES; d = s; }
    float a = xl[s] + xr[d];
    a = (a > 0.f) ? a : NEG_SLOPE * a;
    float ex = __expf(a * att[0] - decodeF(amaxU[d]));
    atomicAdd(denom + d, ex);
    atomicAdd(num + d, ex * xl[s]);
}

/* -------- K4: finalize score and build topk keys for score[USERS:] -------- */
__global__ void k_score(const float* __restrict__ num, const float* __restrict__ denom,
                        const float* __restrict__ bias, float* __restrict__ score,
                        unsigned* __restrict__ keys)
{
    int n = blockIdx.x * blockDim.x + threadIdx.x;
    if (n >= N_NODES) return;
    float sc = num[n] / (denom[n] + 1e-16f) + bias[0];
    score[n] = sc;
    if (n >= USERS) keys[n - USERS] = encodeF(sc);
}

/* -------- K5: radix-select the KSEL-th largest key (single block, LDS histogram) -------- */
__global__ __launch_bounds__(256) void k_radix_select(const unsigned* __restrict__ keys,
                                                      unsigned* __restrict__ ctr)
{
    __shared__ unsigned hist[256];
    __shared__ unsigned sPrefix, sMask, sRemaining;
    const int tid = threadIdx.x;
    if (tid == 0) { sPrefix = 0u; sMask = 0u; sRemaining = KSEL; }
    __syncthreads();
    const int M = N_NODES - USERS;

    for (int pass = 0; pass < 4; ++pass) {
        const int shift = 24 - 8 * pass;
        hist[tid] = 0u;
        __syncthreads();
        const unsigned prefix = sPrefix, mask = sMask;
        for (int i = tid; i < M; i += 256) {
            unsigned key = keys[i];
            if ((key & mask) == prefix)
                atomicAdd(&hist[(key >> shift) & 255u], 1u);
        }
        __syncthreads();
        if (tid == 0) {
            unsigned rem = sRemaining, cum = 0u;
            int dsel = 0;
            for (int d = 255; d >= 0; --d) {
                unsigned cnt = hist[d];
                if (cum + cnt >= rem) { dsel = d; break; }
                cum += cnt;
            }
            sRemaining = rem - cum;
            sPrefix    = prefix | ((unsigned)dsel << shift);
            sMask      = mask | (255u << shift);
        }
        __syncthreads();
    }
    if (tid == 0) {
        ctr[0] = sPrefix;              /* threshold key T */
        ctr[1] = sRemaining;           /* ties to take     */
        ctr[2] = KSEL - sRemaining;    /* count strictly > T */
        ctr[3] = 0u;                   /* gt compaction counter  */
        ctr[4] = 0u;                   /* tie compaction counter */
    }
}

/* -------- K6: compact selected indices into perm[0..KSEL) + user tail -------- */
__global__ void k_compact(const unsigned* __restrict__ keys, unsigned* __restrict__ ctr,
                          int* __restrict__ perm)
{
    int i = blockIdx.x * blockDim.x + threadIdx.x;
    if (i < N_NODES - USERS) {
        unsigned key = keys[i];
        unsigned T = ctr[0];
        if (key > T) {
            unsigned p = atomicAdd(&ctr[3], 1u);
            perm[p] = i;               /* slice-relative index, offset quirk preserved */
        } else if (key == T) {
            unsigned t = atomicAdd(&ctr[4], 1u);
            if (t < ctr[1]) perm[ctr[2] + t] = i;
        }
    }
    if (i < USERS) perm[KSEL + i] = i;
}

/* -------- K7: mapping[perm[j]] = j -------- */
__global__ void k_scatter_map(const int* __restrict__ perm, int* __restrict__ mapping)
{
    int j = blockIdx.x * blockDim.x + threadIdx.x;
    if (j < PERM_LEN) mapping[perm[j]] = j;
}

/* -------- K8: x_new = x[perm] * tanh(score[perm]); batch/perm/n_id outputs -------- */
__global__ void k_gather_x(const int* __restrict__ perm, const float* __restrict__ score,
                           const float* __restrict__ x, const int* __restrict__ batch,
                           const int* __restrict__ n_id, float* __restrict__ out)
{
    long idx = (long)blockIdx.x * blockDim.x + threadIdx.x;
    if (idx >= (long)PERM_LEN * DIM) return;
    int j   = (int)(idx >> 7);
    int col = (int)(idx & 127);
    int p   = perm[j];
    float t = tanhf(score[p]);
    __builtin_nontemporal_store(x[(long)p * DIM + col] * t, out + X_OFF + idx);
    if (col == 0) {
        __builtin_nontemporal_store((float)batch[p], out + BATCH_OFF + j);
        __builtin_nontemporal_store((float)p,        out + PERM_OFF + j);
        __builtin_nontemporal_store((float)n_id[p],  out + NID_OFF + j);
    }
}

/* -------- K9: filter_adj — reindex edges, mask dropped endpoints -------- */
__global__ void k_filter_edges(const int* __restrict__ ei, const float* __restrict__ ea,
                               const int* __restrict__ mapping, float* __restrict__ out)
{
    int e = blockIdx.x * blockDim.x + threadIdx.x;
    if (e >= N_EDGES) return;
    __builtin_prefetch(ei + e + 8192, 0, 1);
    int r = mapping[__builtin_nontemporal_load(ei + e)];
    int c = mapping[__builtin_nontemporal_load(ei + N_EDGES + e)];
    float a = __builtin_nontemporal_load(ea + e);
    bool ok = (r >= 0) && (c >= 0);
    __builtin_nontemporal_store(ok ? (float)r : -1.f, out + EI_OFF + e);
    __builtin_nontemporal_store(ok ? (float)c : -1.f, out + EI_OFF + N_EDGES + e);
    __builtin_nontemporal_store(ok ? a : 0.f,         out + EA_OFF + e);
}

extern "C" void kernel_launch(void* const* d_in, const int* in_sizes, int n_in,
                              void* d_out, int out_size, void* d_ws, size_t ws_size,
                              hipStream_t stream)
{
    (void)in_sizes; (void)n_in; (void)out_size; (void)ws_size;
    const float* x     = (const float*)d_in[0];
    const int*   ei    = (const int*)d_in[1];
    const float* ea    = (const float*)d_in[2];
    const int*   batch = (const int*)d_in[3];
    const int*   n_id  = (const int*)d_in[4];
    const float* Wl    = (const float*)d_in[5];
    const float* bl    = (const float*)d_in[6];
    const float* Wr    = (const float*)d_in[7];
    const float* br    = (const float*)d_in[8];
    const float* att   = (const float*)d_in[9];
    const float* bias  = (const float*)d_in[10];

    float*    wsf = (float*)d_ws;
    unsigned* wsu = (unsigned*)d_ws;
    int*      wsi = (int*)d_ws;
    float*    out = (float*)d_out;

    float*    xl      = wsf + WS_XL;
    float*    xr      = wsf + WS_XR;
    unsigned* amaxU   = wsu + WS_AMAX;
    float*    denom   = wsf + WS_DENOM;
    float*    num     = wsf + WS_NUM;
    float*    score   = wsf + WS_SCORE;
    unsigned* keys    = wsu + WS_KEYS;
    int*      mapping = wsi + WS_MAP;
    int*      perm    = wsi + WS_PERM;
    unsigned* ctr     = wsu + WS_CTR;

    const int EN = N_EDGES + N_NODES;

    k_init        <<<(N_NODES + 255) / 256, 256, 0, stream>>>(amaxU, denom, num, mapping);
    k_gemv_wmma   <<<(N_TILES + 7) / 8,     256, 0, stream>>>(x, Wl, Wr, bl, br, xl, xr);
    k_edge_max    <<<(EN + 255) / 256,      256, 0, stream>>>(ei, xl, xr, att, amaxU);
    k_edge_sum    <<<(EN + 255) / 256,      256, 0, stream>>>(ei, xl, xr, att, amaxU, denom, num);
    k_score       <<<(N_NODES + 255) / 256, 256, 0, stream>>>(num, denom, bias, score, keys);
    k_radix_select<<<1,                     256, 0, stream>>>(keys, ctr);
    k_compact     <<<(N_NODES - USERS + 255) / 256, 256, 0, stream>>>(keys, ctr, perm);
    k_scatter_map <<<(PERM_LEN + 255) / 256, 256, 0, stream>>>(perm, mapping);
    k_gather_x    <<<((long)PERM_LEN * DIM + 255) / 256, 256, 0, stream>>>(perm, score, x, batch, n_id, out);
    k_filter_edges<<<(N_EDGES + 255) / 256, 256, 0, stream>>>(ei, ea, mapping, out);
}